// LDEPooling_74783970558328
// MI455X (gfx1250) — compile-verified
//
#include <hip/hip_runtime.h>

typedef __attribute__((ext_vector_type(2))) float v2f;
typedef __attribute__((ext_vector_type(8))) float v8f;

#define Bn 16
#define Tn 2048
#define Dn 256
#define Cn 8
#define TILE_T 64
#define TILES (Tn / TILE_T)      // 32
#define LDS_STRIDE 260           // 256 + 4 pad floats (mult of 4 -> b128 aligned)
#define CROWS (Cn + 1)           // 8 center rows + 1 shared zero row
#define NTHREADS 128             // 4 waves of 32
#define NWAVES 4

// ---------------------------------------------------------------------------
// Fused LDE pooling, one HBM pass over x per tile.
//  - x tile + zero-padded centers staged in LDS (~80KB -> 4 WGs / WGP)
//  - xc = x @ centers^T via V_WMMA_F32_16X16X4_F32 (per-wave 16 rows, K=256)
//  - softmax over C done IN REGISTERS on the WMMA D layout (c = lane) with
//    16-lane shfl_xor butterflies; padded lanes contribute exp(-inf) = 0
//  - P = assignment^T @ x_tile via WMMA (K=64), waves split the 16 D-tiles
//  - deterministic per-(b,tile) partial stores, no atomics
// ---------------------------------------------------------------------------
__global__ __launch_bounds__(NTHREADS) void lde_main(
    const float* __restrict__ x,        // (B,T,D)
    const float* __restrict__ centers,  // (C,D)
    const float* __restrict__ scale,    // (C,)
    float* __restrict__ ws_agg,         // (B,TILES,Cn,Dn)
    float* __restrict__ ws_asum)        // (B,TILES,Cn)
{
    __shared__ float lds_x[TILE_T * LDS_STRIDE];   // 66,560 B
    __shared__ float lds_c[CROWS * LDS_STRIDE];    //  9,360 B (row 8 = zeros)
    __shared__ float lds_asg[TILE_T * 16];         //  4,096 B  asg[t][c], c>=8 -> 0
    __shared__ float lds_xsq[TILE_T];              //    256 B
    __shared__ float lds_csq[16];
    __shared__ float lds_scale[16];

    const int tid   = threadIdx.x;
    const int wave  = tid >> 5;
    const int lane  = tid & 31;
    const int tileT = blockIdx.x % TILES;
    const int b     = blockIdx.x / TILES;
    const long long xbase = ((long long)b * Tn + (long long)tileT * TILE_T) * Dn;

    // ---- Phase 1: stage x tile (contiguous) + centers (+ zero row), float4
    for (int i = tid; i < TILE_T * (Dn / 4); i += NTHREADS) {
        const int row = i >> 6;        // / (Dn/4)
        const int q   = i & 63;
        const float4 v = ((const float4*)(x + xbase))[i];
        float* dst = &lds_x[row * LDS_STRIDE + q * 4];
        dst[0] = v.x; dst[1] = v.y; dst[2] = v.z; dst[3] = v.w;
    }
    for (int i = tid; i < CROWS * (Dn / 4); i += NTHREADS) {
        const int row = i >> 6;
        const int q   = i & 63;
        float4 v = make_float4(0.f, 0.f, 0.f, 0.f);
        if (row < Cn) v = ((const float4*)centers)[row * (Dn / 4) + q];
        float* dst = &lds_c[row * LDS_STRIDE + q * 4];
        dst[0] = v.x; dst[1] = v.y; dst[2] = v.z; dst[3] = v.w;
    }
    if (tid < 16) lds_scale[tid] = (tid < Cn) ? scale[tid] : 0.f;
    __syncthreads();

    // ---- Phase 2: ||x_row||^2 (threads 0..63); ||c||^2 (threads 64..79)
    if (tid < TILE_T) {
        const float* r = &lds_x[tid * LDS_STRIDE];
        float s = 0.f;
        for (int d = 0; d < Dn; d += 4) {
            const float4 v = *(const float4*)(r + d);
            s += v.x * v.x + v.y * v.y + v.z * v.z + v.w * v.w;
        }
        lds_xsq[tid] = s;
    } else if (tid < TILE_T + 16) {
        const int c = tid - TILE_T;
        const float* r = &lds_c[(c < Cn ? c : Cn) * LDS_STRIDE];  // zeros if c>=8
        float s = 0.f;
        for (int d = 0; d < Dn; d += 4) {
            const float4 v = *(const float4*)(r + d);
            s += v.x * v.x + v.y * v.y + v.z * v.z + v.w * v.w;
        }
        lds_csq[c] = s;
    }
    __syncthreads();

    // ---- Phase 3: xc via f32 WMMA + in-register softmax. Wave: rows wave*16..+15
    // A layout (16x4 f32): m = lane&15, k = 2*(lane>>4) + v
    // B layout (4x16 f32): n = lane&15, k = 2*(lane>>4) + v ; n>=8 -> zero row
    {
        v8f acc = {};
        const int m     = lane & 15;
        const int khalf = (lane >> 4) << 1;
        const float* arow = &lds_x[(wave * 16 + m) * LDS_STRIDE];
        const float* crow = &lds_c[(m < Cn ? m : Cn) * LDS_STRIDE];
        for (int kc = 0; kc < Dn / 4; ++kc) {
            const int k0 = kc * 4 + khalf;
            v2f a;  a.x  = arow[k0]; a.y  = arow[k0 + 1];
            v2f bb; bb.x = crow[k0]; bb.y = crow[k0 + 1];
            acc = __builtin_amdgcn_wmma_f32_16x16x4_f32(
                false, a, false, bb, (short)0, acc, false, false);
        }
        // D layout: lane l, vgpr r -> row M = r + 8*(l>>4), center c = l&15.
        // Softmax over c == butterfly across each 16-lane half.
        const int   c     = lane & 15;
        const int   Mbase = (lane >> 4) * 8;
        const float csq   = lds_csq[c];
        const float scl   = lds_scale[c];
        const bool  real  = (c < Cn);
        float aout[8];
        #pragma unroll
        for (int r = 0; r < 8; ++r) {
            const float xsq = lds_xsq[wave * 16 + Mbase + r];
            float val = -scl * (xsq - 2.f * acc[r] + csq);
            val = real ? val : -1e30f;
            float mx = val;
            #pragma unroll
            for (int msk = 1; msk <= 8; msk <<= 1)
                mx = fmaxf(mx, __shfl_xor(mx, msk, 32));
            const float e = __expf(val - mx);
            float sum = e;
            #pragma unroll
            for (int msk = 1; msk <= 8; msk <<= 1)
                sum += __shfl_xor(sum, msk, 32);
            aout[r] = e / sum;      // exact 0 for c>=8
        }
        #pragma unroll
        for (int r = 0; r < 8; ++r)
            lds_asg[(wave * 16 + Mbase + r) * 16 + c] = aout[r];
    }
    __syncthreads();

    // ---- Phase 4: P[c, d] = sum_t asg[t][c] * x[t][d], K = TILE_T rows.
    // Waves split the 16 D-tiles: wave owns dtiles 4*wave .. 4*wave+3.
    {
        const int m     = lane & 15;            // A: c index / B: n column
        const int khalf = (lane >> 4) << 1;
        for (int dt = wave * 4; dt < wave * 4 + 4; ++dt) {
            v8f acc = {};
            for (int kc = 0; kc < TILE_T / 4; ++kc) {
                const int k0 = kc * 4 + khalf;
                v2f a;  a.x  = lds_asg[k0 * 16 + m];
                        a.y  = lds_asg[(k0 + 1) * 16 + m];
                v2f bb; bb.x = lds_x[k0 * LDS_STRIDE + dt * 16 + m];
                        bb.y = lds_x[(k0 + 1) * LDS_STRIDE + dt * 16 + m];
                acc = __builtin_amdgcn_wmma_f32_16x16x4_f32(
                    false, a, false, bb, (short)0, acc, false, false);
            }
            // Only M = c < 8 matters: lanes 0..15 hold M = r (r = vgpr index)
            if (lane < 16) {
                float* dst = &ws_agg[((long long)(b * TILES + tileT) * Cn) * Dn
                                     + dt * 16 + lane];
                #pragma unroll
                for (int r = 0; r < Cn; ++r) dst[(long long)r * Dn] = acc[r];
            }
        }
    }

    // ---- a_sum partial for this tile (wave 0, lanes 0..7), fixed order
    if (wave == 0 && lane < Cn) {
        float s = 0.f;
        for (int t2 = 0; t2 < TILE_T; ++t2) s += lds_asg[t2 * 16 + lane];
        ws_asum[(b * TILES + tileT) * Cn + lane] = s;
    }
}

// ---------------------------------------------------------------------------
// Finalize: reduce T-tiles in fixed order, apply agg - a_sum * centers.
// ---------------------------------------------------------------------------
__global__ __launch_bounds__(256) void lde_finalize(
    const float* __restrict__ ws_agg,
    const float* __restrict__ ws_asum,
    const float* __restrict__ centers,
    float* __restrict__ out)
{
    const int idx = blockIdx.x * 256 + threadIdx.x;   // over B*Cn*Dn
    if (idx >= Bn * Cn * Dn) return;
    const int d = idx & (Dn - 1);
    const int c = (idx >> 8) & (Cn - 1);
    const int b = idx >> 11;
    float agg = 0.f, asum = 0.f;
    for (int t = 0; t < TILES; ++t) {
        agg  += ws_agg[((long long)(b * TILES + t) * Cn + c) * Dn + d];
        asum += ws_asum[(b * TILES + t) * Cn + c];
    }
    out[idx] = agg - asum * centers[c * Dn + d];
}

// ---------------------------------------------------------------------------
extern "C" void kernel_launch(void* const* d_in, const int* in_sizes, int n_in,
                              void* d_out, int out_size, void* d_ws, size_t ws_size,
                              hipStream_t stream)
{
    const float* x       = (const float*)d_in[0];
    const float* centers = (const float*)d_in[1];
    const float* scale   = (const float*)d_in[2];
    float* out = (float*)d_out;

    float* ws_agg  = (float*)d_ws;                                   // B*TILES*Cn*Dn
    float* ws_asum = ws_agg + (size_t)Bn * TILES * Cn * Dn;          // B*TILES*Cn

    lde_main<<<Bn * TILES, NTHREADS, 0, stream>>>(x, centers, scale, ws_agg, ws_asum);
    lde_finalize<<<(Bn * Cn * Dn + 255) / 256, 256, 0, stream>>>(
        ws_agg, ws_asum, centers, out);
}